// LavaNetwork_49941879718260
// MI455X (gfx1250) — compile-verified
//
#include <hip/hip_runtime.h>

// Problem constants (from reference setup_inputs)
#define B_SZ  64
#define T_SZ  2048
#define NIN   96
#define NHID  256
#define NOUT  80

typedef __attribute__((ext_vector_type(2))) float v2f;
typedef __attribute__((ext_vector_type(8))) float v8f;

// ---------------------------------------------------------------------------
// Time-batched GEMM via V_WMMA_F32_16X16X4_F32 (exact f32 math).
//   z[tl, b, n] = sum_k act[b, k, t0+tl] * W[n, k]
// One wave32 per (b, NT*16-column group, MT m-tiles). The W fragments for all
// K-steps are preloaded into registers once and reused across all MT timestep
// tiles; each A fragment is reused across NT column tiles.
// A layout (32-bit A 16x4): lanes 0-15 -> K = k0,k0+1 ; lanes 16-31 -> k0+2,k0+3
// B layout mirrored; C/D: VGPR r holds rows M=r (lanes 0-15) / M=r+8 (16-31).
// ---------------------------------------------------------------------------
template <int K, int NT>
__global__ __launch_bounds__(32) void cuba_gemm_wmma(
    const float* __restrict__ act,   // [B, K, T_SZ]
    const float* __restrict__ W,     // [N, K]
    float* __restrict__ z,           // [Tc, B, N] chunk buffer
    int N, int t0, int MT, int msplit) {
  const int lane = threadIdx.x & 31;
  const int half = lane >> 4;        // which K-pair this half-wave supplies
  const int l    = lane & 15;        // row (A: t) / col (B: n) within tile
  const int b    = blockIdx.x / msplit;
  const int ms   = blockIdx.x - b * msplit;
  const int n0   = blockIdx.y * (16 * NT);

  // Preload the whole W tile (NT columns x K) as WMMA B-fragments in VGPRs.
  v2f wf[K / 4][NT];
#pragma unroll
  for (int s = 0; s < K / 4; ++s) {
    const int ka = 4 * s + 2 * half;
#pragma unroll
    for (int nt = 0; nt < NT; ++nt) {
      const float* wrow = W + (size_t)(n0 + nt * 16 + l) * K;
      wf[s][nt][0] = wrow[ka];
      wf[s][nt][1] = wrow[ka + 1];
    }
  }

  const size_t rowstride = (size_t)B_SZ * N;
  const int mt_end = ms * MT + MT;
  for (int mt = ms * MT; mt < mt_end; ++mt) {
    const float* arow = act + (size_t)b * K * T_SZ + (t0 + mt * 16 + l);
    v8f c[NT] = {};
#pragma unroll
    for (int s = 0; s < K / 4; ++s) {
      const int ka = 4 * s + 2 * half;
      v2f a = { arow[(size_t)ka * T_SZ], arow[(size_t)(ka + 1) * T_SZ] };
#pragma unroll
      for (int nt = 0; nt < NT; ++nt)
        c[nt] = __builtin_amdgcn_wmma_f32_16x16x4_f32(
            false, a, false, wf[s][nt], (short)0, c[nt], false, false);
    }
    float* zbase = z + (size_t)(mt * 16 + 8 * half) * rowstride
                     + (size_t)b * N + l;
#pragma unroll
    for (int nt = 0; nt < NT; ++nt)
#pragma unroll
      for (int r = 0; r < 8; ++r)
        zbase[(size_t)r * rowstride + n0 + nt * 16] = c[nt][r];
  }
}

// ---------------------------------------------------------------------------
// CUBA-LIF recurrence over one T-chunk. One thread per (b, neuron); z is read
// coalesced across neurons, spikes stream out as float4 per thread.
//   v = 0.95*v + z;  s = (v >= 1);  v *= (1 - s)
// ---------------------------------------------------------------------------
__global__ __launch_bounds__(256) void cuba_lif_scan(
    const float* __restrict__ z,      // [Tc, B, N]
    float* __restrict__ spk,          // [B, N, T_SZ]
    float* __restrict__ vstate,       // [B*N] carried across chunks
    int N, int t0, int Tc, int init) {
  const int tid   = blockIdx.x * 256 + threadIdx.x;   // = b*N + o
  const int total = B_SZ * N;
  if (tid >= total) return;

  float v = init ? 0.0f : vstate[tid];
  const float* zp = z + tid;
  float* out = spk + (size_t)tid * T_SZ + t0;

  float4 buf;
  for (int tl = 0; tl < Tc; ++tl) {
    const float cur = zp[(size_t)tl * total];
    v = v * 0.95f + cur;                 // leaky integrate (beta = 1 - V_DECAY)
    const float s = (v >= 1.0f) ? 1.0f : 0.0f;
    v -= v * s;                          // hard reset on spike
    (&buf.x)[tl & 3] = s;
    if ((tl & 3) == 3) *(float4*)(out + (tl - 3)) = buf;
  }
  vstate[tid] = v;
}

// ---------------------------------------------------------------------------
// Per chunk (Tc timesteps): gemm1 -> scan1 -> gemm2 -> scan2, all in-stream so
// the z slab (<=16.8 MB) stays L2-resident between producer and consumer.
// ---------------------------------------------------------------------------
extern "C" void kernel_launch(void* const* d_in, const int* in_sizes, int n_in,
                              void* d_out, int out_size, void* d_ws, size_t ws_size,
                              hipStream_t stream) {
  const float* x  = (const float*)d_in[0];   // [B, NIN, T]
  const float* W1 = (const float*)d_in[1];   // [NHID, NIN]
  const float* W2 = (const float*)d_in[2];   // [NOUT, NHID]

  float* spk1 = (float*)d_out;                           // [B, NHID, T]
  float* spk2 = spk1 + (size_t)B_SZ * NHID * T_SZ;       // [B, NOUT, T]

  // Pick largest T-chunk whose z slab + state vectors fit in the workspace.
  int Tc = 256;
  while (Tc > 16) {
    size_t need = ((size_t)Tc * B_SZ * NHID + (size_t)B_SZ * NHID
                   + (size_t)B_SZ * NOUT) * sizeof(float);
    if (need <= ws_size) break;
    Tc >>= 1;
  }
  const int mtiles = Tc / 16;
  const int msplit = mtiles < 4 ? mtiles : 4;   // waves per chunk in m-dim
  const int MT     = mtiles / msplit;           // m-tiles per wave

  float* zbuf = (float*)d_ws;                            // [Tc, B, Nmax]
  float* v1   = zbuf + (size_t)Tc * B_SZ * NHID;         // [B*NHID]
  float* v2   = v1 + (size_t)B_SZ * NHID;                // [B*NOUT]

  for (int t0 = 0; t0 < T_SZ; t0 += Tc) {
    const int init = (t0 == 0);
    cuba_gemm_wmma<NIN, 2><<<dim3(B_SZ * msplit, (NHID / 16) / 2), 32, 0, stream>>>(
        x, W1, zbuf, NHID, t0, MT, msplit);
    cuba_lif_scan<<<(B_SZ * NHID) / 256, 256, 0, stream>>>(
        zbuf, spk1, v1, NHID, t0, Tc, init);
    cuba_gemm_wmma<NHID, 1><<<dim3(B_SZ * msplit, NOUT / 16), 32, 0, stream>>>(
        spk1, W2, zbuf, NOUT, t0, MT, msplit);
    cuba_lif_scan<<<(B_SZ * NOUT) / 256, 256, 0, stream>>>(
        zbuf, spk2, v2, NOUT, t0, Tc, init);
  }
}